// MetaGCN_8194797601143
// MI455X (gfx1250) — compile-verified
//
#include <hip/hip_runtime.h>
#include <hip/hip_bf16.h>

// MetaGCN on MI455X (gfx1250, wave32).
//
// Roofline: GEMM is 2.5 GFLOP (noise). The edge stage dominates and is
// atomic-limited, not bandwidth-limited (total HBM traffic ~300MB ≈ 13us at
// 23.3 TB/s). So: build a destination-CSR on device (9M cheap atomics) and do
// a per-node register reduction with one coalesced write per output line,
// instead of 192M global f32 atomics. h/out slabs (25.6MB each) are L2-hot
// (192MB L2). GEMM uses V_WMMA_F32_16X16X4_F32 (fp32-exact), with W[mp]
// staged into LDS by the Tensor Data Mover (TENSOR_LOAD_TO_LDS + s_wait_tensorcnt).

typedef __attribute__((ext_vector_type(2))) float    v2f;
typedef __attribute__((ext_vector_type(8))) float    v8f;
typedef __attribute__((ext_vector_type(4))) unsigned u32x4;
typedef __attribute__((ext_vector_type(8))) int      i32x8;
typedef __attribute__((ext_vector_type(4))) int      i32x4;

#define NN   100000        // nodes
#define NMP  3             // metapaths
#define NE   1000000       // edges per metapath
#define D    64            // feature dim
#define MT   (NN / 16)     // 6250 row tiles (exact)
#define NB   (NMP * NN)    // 300000 destination bins
#define SBLK 256
#define NBLK ((NB + SBLK - 1) / SBLK)   // 1172

// ---------------- zero fill (grid-stride) ----------------
__global__ void MetaGCN_zero_kernel(float* __restrict__ p, size_t n) {
    size_t stride = (size_t)gridDim.x * blockDim.x;
    for (size_t i = (size_t)blockIdx.x * blockDim.x + threadIdx.x; i < n; i += stride)
        p[i] = 0.0f;
}

// ---------------- deg[bin] += ew ; cnt[bin] += 1 ----------------
__global__ void MetaGCN_degcnt_kernel(const long long* __restrict__ ei,
                                      const float* __restrict__ ew,
                                      float* __restrict__ deg,
                                      unsigned* __restrict__ cnt) {
    size_t t = (size_t)blockIdx.x * blockDim.x + threadIdx.x;
    if (t >= (size_t)NMP * NE) return;
    int mp = (int)(t / NE);
    size_t e = t % NE;
    long long col = ei[(size_t)mp * 2 * NE + NE + e];
    unsigned bin = (unsigned)mp * NN + (unsigned)col;
    atomicAdd(&deg[bin], ew[(size_t)mp * NE + e]);
    atomicAdd(&cnt[bin], 1u);
}

// ---------------- dinv = deg>0 ? rsqrt(deg) : 0 (in place) ----------------
__global__ void MetaGCN_dinv_kernel(float* __restrict__ deg) {
    size_t i = (size_t)blockIdx.x * blockDim.x + threadIdx.x;
    if (i >= (size_t)NB) return;
    float d = deg[i];
    deg[i] = (d > 0.0f) ? rsqrtf(d) : 0.0f;
}

// ---------------- scan stage 1: per-block sums of cnt ----------------
__global__ void MetaGCN_bsum_kernel(const unsigned* __restrict__ cnt,
                                    unsigned* __restrict__ bsum) {
    __shared__ unsigned s[SBLK];
    int i = blockIdx.x * SBLK + threadIdx.x;
    s[threadIdx.x] = (i < NB) ? cnt[i] : 0u;
    __syncthreads();
    for (int o = SBLK / 2; o > 0; o >>= 1) {
        if ((int)threadIdx.x < o) s[threadIdx.x] += s[threadIdx.x + o];
        __syncthreads();
    }
    if (threadIdx.x == 0) bsum[blockIdx.x] = s[0];
}

// ---------------- scan stage 2: serial exclusive scan of 1172 block sums ----------------
__global__ void MetaGCN_scanb_kernel(unsigned* __restrict__ bsum, int n) {
    unsigned acc = 0;
    for (int i = 0; i < n; ++i) { unsigned v = bsum[i]; bsum[i] = acc; acc += v; }
}

// ---------------- scan stage 3: LDS Hillis-Steele, global exclusive offsets ----------------
__global__ void MetaGCN_scanf_kernel(const unsigned* __restrict__ cnt,
                                     const unsigned* __restrict__ bsum,
                                     unsigned* __restrict__ ofs,
                                     unsigned* __restrict__ epos) {
    __shared__ unsigned s[SBLK];
    int i = blockIdx.x * SBLK + threadIdx.x;
    unsigned v = (i < NB) ? cnt[i] : 0u;
    s[threadIdx.x] = v;
    __syncthreads();
    for (int o = 1; o < SBLK; o <<= 1) {
        unsigned t = (threadIdx.x >= (unsigned)o) ? s[threadIdx.x - o] : 0u;
        __syncthreads();
        s[threadIdx.x] += t;
        __syncthreads();
    }
    unsigned ex = s[threadIdx.x] - v + bsum[blockIdx.x];
    if (i < NB) { ofs[i] = ex; epos[i] = ex; }
}

// ---------------- CSR fill: place edge ids by destination bin ----------------
__global__ void MetaGCN_fill_kernel(const long long* __restrict__ ei,
                                    unsigned* __restrict__ epos,
                                    unsigned* __restrict__ eidx) {
    size_t t = (size_t)blockIdx.x * blockDim.x + threadIdx.x;
    if (t >= (size_t)NMP * NE) return;
    int mp = (int)(t / NE);
    unsigned e = (unsigned)(t % NE);
    long long col = ei[(size_t)mp * 2 * NE + NE + e];
    unsigned bin = (unsigned)mp * NN + (unsigned)col;
    unsigned p = atomicAdd(&epos[bin], 1u);
    eidx[p] = e;
}

// ---------------- h[mp] = x @ W[mp] : TDM-staged W + V_WMMA_F32_16X16X4_F32 ----
// Block = 320 threads = 10 waves, each wave owns a 16x64 output row strip
// (4 accumulator tiles, 64 WMMAs, 4x A reuse). 6250 strips/mp = 625 blocks/mp
// exactly, so mp is block-uniform -> uniform SGPR tensor descriptor.
// W[mp] (64x64 f32 = 16KB) is DMA'd to LDS by wave 0 via TENSOR_LOAD_TO_LDS.
__global__ void __launch_bounds__(320)
MetaGCN_gemm_wmma_kernel(const float* __restrict__ x,
                         const float* __restrict__ W,
                         float* __restrict__ h) {
    __shared__ float lW[D * D];

    int mp     = blockIdx.x / 625;
    int tmbase = (blockIdx.x % 625) * 10;

    if (threadIdx.x < 32) {
        // Tensor DMA descriptor (D#): 2D load, data_size=4B, 64x64 tile,
        // tensor_dim0=64, dim0_stride=64, dest = lW.
        unsigned long long ga = (unsigned long long)(uintptr_t)(W + (size_t)mp * D * D);
        unsigned ldsoff = (unsigned)(uintptr_t)(&lW[0]);   // addr[31:0] = LDS offset
        u32x4 g0;
        g0[0] = 1u;                                        // count=1, user mode
        g0[1] = ldsoff;                                    // lds_addr
        g0[2] = (unsigned)(ga & 0xffffffffu);              // global_addr[31:0]
        g0[3] = ((unsigned)(ga >> 32) & 0x01ffffffu) | (2u << 30);  // addr[56:32], type=2
        i32x8 g1;
        g1[0] = 0x00020000;          // data_size=2 (4B), no multicast/pad/iterate
        g1[1] = (int)(64u << 16);    // tensor_dim0[15:0]=64
        g1[2] = (int)(64u << 16);    // tensor_dim0[31:16]=0, tensor_dim1[15:0]=64
        g1[3] = (int)(64u << 16);    // tensor_dim1[31:16]=0, tile_dim0=64
        g1[4] = 0x00000040;          // tile_dim1=64, tile_dim2=0
        g1[5] = 64;                  // tensor_dim0_stride[31:0]=64
        g1[6] = 0;                   // stride[47:32]=0, dim1_stride lo=0
        g1[7] = 0;
        i32x4 z4 = {0, 0, 0, 0};
#if defined(__clang_major__) && (__clang_major__ >= 23)
        i32x8 z8 = {0, 0, 0, 0, 0, 0, 0, 0};
        __builtin_amdgcn_tensor_load_to_lds(g0, g1, z4, z4, z8, 0);
#else
        __builtin_amdgcn_tensor_load_to_lds(g0, g1, z4, z4, 0);
#endif
        __builtin_amdgcn_s_wait_tensorcnt(0);
    }
    __syncthreads();

    int wv   = threadIdx.x >> 5;      // 0..9
    int tm   = tmbase + wv;
    int lane = threadIdx.x & 31;
    int half = lane >> 4;             // K pair select
    int lid  = lane & 15;

    const float* ap = x + (size_t)(tm * 16 + lid) * D + half * 2;

    v8f c0 = {}, c1 = {}, c2 = {}, c3 = {};
#pragma unroll
    for (int kk = 0; kk < D; kk += 4) {
        v2f a;
        a[0] = ap[0];
        a[1] = ap[1];
        const float* bp = &lW[(kk + half * 2) * D + lid];   // row kk+2*half, col lid
        v2f b;
        b[0] = bp[0];   b[1] = bp[D];        // cols 0..15
        c0 = __builtin_amdgcn_wmma_f32_16x16x4_f32(false, a, false, b, (short)0, c0, false, false);
        b[0] = bp[16];  b[1] = bp[D + 16];   // cols 16..31
        c1 = __builtin_amdgcn_wmma_f32_16x16x4_f32(false, a, false, b, (short)0, c1, false, false);
        b[0] = bp[32];  b[1] = bp[D + 32];   // cols 32..47
        c2 = __builtin_amdgcn_wmma_f32_16x16x4_f32(false, a, false, b, (short)0, c2, false, false);
        b[0] = bp[48];  b[1] = bp[D + 48];   // cols 48..63
        c3 = __builtin_amdgcn_wmma_f32_16x16x4_f32(false, a, false, b, (short)0, c3, false, false);
        ap += 4;
    }

    float* hp = h + ((size_t)mp * NN + (size_t)(tm * 16 + half * 8)) * D + lid;
#pragma unroll
    for (int r = 0; r < 8; ++r) {
        hp[(size_t)r * D +  0] = c0[r];
        hp[(size_t)r * D + 16] = c1[r];
        hp[(size_t)r * D + 32] = c2[r];
        hp[(size_t)r * D + 48] = c3[r];
    }
}

// ---------------- gather-reduce: one wave per (mp,node), fused ReLU ----------
// Replaces 192M global f32 atomics with register accumulation + one coalesced
// 256B write per node. h gathers hit L2 (25.6MB slab per metapath).
__global__ void MetaGCN_gather_kernel(const long long* __restrict__ ei,
                                      const float* __restrict__ ew,
                                      const float* __restrict__ dinv,
                                      const unsigned* __restrict__ ofs,
                                      const unsigned* __restrict__ cnt,
                                      const unsigned* __restrict__ eidx,
                                      const float* __restrict__ h,
                                      float* __restrict__ out) {
    int wave = blockIdx.x * (blockDim.x >> 5) + (threadIdx.x >> 5);
    int lane = threadIdx.x & 31;
    if (wave >= NB) return;
    int mp   = wave / NN;
    unsigned start = ofs[wave];
    unsigned n     = cnt[wave];
    float dc = dinv[wave];

    const long long* rowp = ei + (size_t)mp * 2 * NE;   // row indices of this mp
    const float*     ewp  = ew + (size_t)mp * NE;
    const float*     dvp  = dinv + (size_t)mp * NN;
    const float*     hb   = h + (size_t)mp * NN * D;

    float a0 = 0.0f, a1 = 0.0f;
    for (unsigned j = 0; j < n; ++j) {
        unsigned e  = eidx[start + j];      // broadcast load (same addr all lanes)
        long long r = rowp[e];
        float nrm   = dvp[r] * ewp[e] * dc;
        const float* hr = hb + (size_t)r * D;
        a0 += hr[lane]      * nrm;
        a1 += hr[lane + 32] * nrm;
    }
    float* op = out + (size_t)wave * D;
    op[lane]      = fmaxf(a0, 0.0f);
    op[lane + 32] = fmaxf(a1, 0.0f);
}

extern "C" void kernel_launch(void* const* d_in, const int* in_sizes, int n_in,
                              void* d_out, int out_size, void* d_ws, size_t ws_size,
                              hipStream_t stream) {
    const float*     x  = (const float*)d_in[0];      // [NN, D]
    const float*     W  = (const float*)d_in[1];      // [NMP, D, D]
    const long long* ei = (const long long*)d_in[2];  // [NMP, 2, NE] int64
    const float*     ew = (const float*)d_in[3];      // [NMP, NE]
    float* out = (float*)d_out;                       // [NMP, NN, D]

    // workspace layout (~94MB)
    float*    deg  = (float*)d_ws;                         // NB floats (-> dinv)
    unsigned* cnt  = (unsigned*)(deg + NB);                // NB
    unsigned* ofs  = cnt + NB;                             // NB
    unsigned* epos = ofs + NB;                             // NB
    unsigned* bsum = epos + NB;                            // NBLK (padded 2048)
    unsigned* eidx = bsum + 2048;                          // NMP*NE
    float*    h    = (float*)(eidx + (size_t)NMP * NE);    // NMP*NN*D floats

    // 1) zero deg + cnt (contiguous 2*NB words)
    MetaGCN_zero_kernel<<<2048, 256, 0, stream>>>(deg, (size_t)2 * NB);

    // 2) degree + histogram: one thread per (mp, edge)
    {
        int blocks = (int)(((size_t)NMP * NE + 255) / 256);
        MetaGCN_degcnt_kernel<<<blocks, 256, 0, stream>>>(ei, ew, deg, cnt);
    }

    // 3) dinv = rsqrt(deg)
    MetaGCN_dinv_kernel<<<(NB + 255) / 256, 256, 0, stream>>>(deg);

    // 4) exclusive scan of cnt -> ofs/epos (3 stages)
    MetaGCN_bsum_kernel<<<NBLK, SBLK, 0, stream>>>(cnt, bsum);
    MetaGCN_scanb_kernel<<<1, 1, 0, stream>>>(bsum, NBLK);
    MetaGCN_scanf_kernel<<<NBLK, SBLK, 0, stream>>>(cnt, bsum, ofs, epos);

    // 5) CSR fill
    {
        int blocks = (int)(((size_t)NMP * NE + 255) / 256);
        MetaGCN_fill_kernel<<<blocks, 256, 0, stream>>>(ei, epos, eidx);
    }

    // 6) WMMA GEMM: 3*625 blocks, 10 waves each, TDM-staged W in LDS
    MetaGCN_gemm_wmma_kernel<<<NMP * 625, 320, 0, stream>>>(x, W, h);

    // 7) per-node gather-reduce with fused ReLU: 300K waves, 8 per block
    MetaGCN_gather_kernel<<<(NB + 7) / 8, 256, 0, stream>>>(ei, ew, deg, ofs, cnt,
                                                            eidx, h, out);
}